// SymFeats_76063870812741
// MI455X (gfx1250) — compile-verified
//
#include <hip/hip_runtime.h>

typedef __attribute__((ext_vector_type(2))) float v2f;
typedef __attribute__((ext_vector_type(8))) float v8f;

#define NA 48
#define NBATCH 32
#define NPAIR 10
#define NFEAT 384
#define PI_F 3.14159265358979323846f

__global__ __launch_bounds__(128)
void aev_kernel(const int* __restrict__ species,
                const float* __restrict__ coords,
                int* __restrict__ sp_out,
                float* __restrict__ aev_out)
{
    const int i    = blockIdx.x;   // center atom 0..47
    const int b    = blockIdx.y;   // batch 0..31
    const int tid  = threadIdx.x;  // 0..127
    const int lane = tid & 31;
    const int wave = tid >> 5;

    __shared__ float xs[NA], ys[NA], zs[NA];
    __shared__ __align__(16) float dpack[NA][4];   // {dx, dy, dz, 0} per neighbor
    __shared__ float dist[NA], fcr[NA], fca[NA];
    __shared__ int   sp[NA];
    __shared__ float D[NA * NA];                   // Gram matrix of displacement vectors
    __shared__ float part[4][32][NPAIR];           // angular partials [chunk][feature][pair]

    // ---- Phase 0: stage batch coords + species into LDS ----
    if (tid < NA) {
        int j = tid;
        xs[j] = coords[(b * NA + j) * 3 + 0];
        ys[j] = coords[(b * NA + j) * 3 + 1];
        zs[j] = coords[(b * NA + j) * 3 + 2];
        sp[j] = species[b * NA + j];
    }
    __syncthreads();

    // ---- Phase 1: diffs (packed), distances, cutoff functions ----
    if (tid < NA) {
        int j = tid;
        float ddx = xs[j] - xs[i];
        float ddy = ys[j] - ys[i];
        float ddz = zs[j] - zs[i];
        dpack[j][0] = ddx;
        dpack[j][1] = ddy;
        dpack[j][2] = ddz;
        dpack[j][3] = 0.0f;                       // K=3 zero pad for the 16x4 WMMA
        float d2 = ddx * ddx + ddy * ddy + ddz * ddz;
        float d  = (j == i) ? 1.0f : sqrtf(d2);   // matches where(eye, 1.0, d2)
        dist[j] = d;
        float fr = 0.0f, fa = 0.0f;
        if (j != i) {
            if (d <= 5.2f) fr = 0.5f * cosf((PI_F * d) / 5.2f) + 0.5f;
            if (d <= 3.5f) fa = 0.5f * cosf((PI_F * d) / 3.5f) + 0.5f;
        }
        fcr[j] = fr;
        fca[j] = fa;
    }
    if (tid == 0) sp_out[b * NA + i] = species[b * NA + i];  // first tuple output
    __syncthreads();

    // ---- Phase 2 (concurrent): wave0 -> Gram matrix via WMMA; waves1-2 -> radial ----
    if (wave == 0) {
        // D[j][k] = diff_j . diff_k  via V_WMMA_F32_16X16X4_F32 (K=4, 4th comp = 0).
        // 16x4 f32 A layout: lanes 0-15 hold K=0,1 ; lanes 16-31 hold K=2,3 (B mirrored).
        // Branchless fragment build: one ds_load_b64 at lane-selected offset into dpack.
        const int fragoff = (lane < 16) ? 0 : 2;     // floats: (dx,dy) vs (dz,0)
        #pragma unroll
        for (int tm = 0; tm < 3; ++tm) {
            int m = tm * 16 + (lane & 15);
            v2f av = *(const v2f*)&dpack[m][fragoff];
            #pragma unroll
            for (int tn = 0; tn < 3; ++tn) {
                int n = tn * 16 + (lane & 15);
                v2f bv = *(const v2f*)&dpack[n][fragoff];
                v8f cacc = {};
                cacc = __builtin_amdgcn_wmma_f32_16x16x4_f32(
                    false, av, false, bv, (short)0, cacc, false, false);
                // C layout: VGPR v, lanes 0-15 -> M=v ; lanes 16-31 -> M=v+8 ; N = lane%16
                int col   = tn * 16 + (lane & 15);
                int rbase = tm * 16 + ((lane < 16) ? 0 : 8);
                #pragma unroll
                for (int v = 0; v < 8; ++v)
                    D[(rbase + v) * NA + col] = cacc[v];
            }
        }
    } else if (wave <= 2) {
        // radial features: one (species s, shift r) per thread, exact f32 sum over j
        int f = tid - 32;                 // 0..63
        int s = f >> 4, r = f & 15;
        float shf = (float)(0.9 + 0.26875 * (double)r);
        float acc = 0.0f;
        for (int j = 0; j < NA; ++j) {
            if (sp[j] == s) {             // one-hot contraction as exact masked sum
                float t = dist[j] - shf;
                acc += 0.25f * expf(-16.0f * t * t) * fcr[j];  // fcr==0 masks eye/cutoff
            }
        }
        aev_out[(b * NA + i) * NFEAT + s * 16 + r] = acc;
    }

    // zero angular partials: each thread owns part[wave][lane][*] exclusively
    for (int p = 0; p < NPAIR; ++p) part[wave][lane][p] = 0.0f;
    __syncthreads();   // D and partials ready

    // ---- Phase 3: angular. 32 features x 4 pair-chunks; lane = feature, wave = chunk ----
    {
        const int f  = lane;              // (a,z) feature
        const int c  = wave;              // chunk of j-rows
        const int ai = f >> 3, zi = f & 7;
        double shz = (2.0 * (double)zi + 1.0) * 3.14159265358979323846 / 16.0;
        const float cz_c = (float)cos(shz);
        const float sz_c = (float)sin(shz);
        const float shfa = (float)(0.9 + 0.65 * (double)ai);
        for (int j = c; j < NA - 1; j += 4) {      // chunked rows, fixed order -> deterministic
            float fj = fca[j];
            if (fj == 0.0f) continue;              // whole row contributes exactly 0
            float dj = dist[j];
            int   sj = sp[j];
            for (int k = j + 1; k < NA; ++k) {
                float fk = fca[k];
                if (fk == 0.0f) continue;          // w==0 -> exact 0 contribution
                float w    = 2.0f * fj * fk;
                float dk   = dist[k];
                float dot  = D[j * NA + k];
                float cosT = 0.95f * dot / (dj * dk);
                float sinT = sqrtf(fmaxf(1.0f - cosT * cosT, 0.0f)); // sin(arccos x)
                float cmix = cosT * cz_c + sinT * sz_c;              // cos(theta - shf_z)
                float base = 0.5f + 0.5f * cmix;
                float p2 = base * base, p4 = p2 * p2, p8 = p4 * p4, p16 = p8 * p8;
                float f1 = p16 * p16;                                // ^32
                float av = 0.5f * (dj + dk) - shfa;
                float f2v = expf(-8.0f * av * av);
                int sk2 = sp[k];
                int s1 = sj < sk2 ? sj : sk2;
                int s2 = sj < sk2 ? sk2 : sj;
                int pidx = ((s1 * (9 - s1)) >> 1) + (s2 - s1);       // symmetric pair index
                part[c][f][pidx] += w * f2v * f1;   // private LDS slice, no races
            }
        }
    }
    __syncthreads();

    // ---- Phase 4: reduce 4 chunks, write angular features ----
    if (tid < 32) {
        int f  = tid;
        int ai = f >> 3, zi = f & 7;
        const int base = (b * NA + i) * NFEAT + 64;
        for (int p = 0; p < NPAIR; ++p) {
            float s = part[0][f][p] + part[1][f][p] + part[2][f][p] + part[3][f][p];
            aev_out[base + p * 32 + ai * 8 + zi] = s;   // layout p*32 + a*8 + z
        }
    }
}

extern "C" void kernel_launch(void* const* d_in, const int* in_sizes, int n_in,
                              void* d_out, int out_size, void* d_ws, size_t ws_size,
                              hipStream_t stream) {
    (void)in_sizes; (void)n_in; (void)out_size; (void)d_ws; (void)ws_size;
    const int*   species = (const int*)d_in[0];
    const float* coords  = (const float*)d_in[1];
    int*   sp_out  = (int*)d_out;                       // tuple output 0: species (1536 i32)
    float* aev_out = (float*)d_out + NBATCH * NA;       // tuple output 1: aevs (1536x384 f32)
    dim3 grid(NA, NBATCH);
    dim3 block(128);
    hipLaunchKernelGGL(aev_kernel, grid, block, 0, stream,
                       species, coords, sp_out, aev_out);
}